// DisentangledStateEncoderCustom_8211977470248
// MI455X (gfx1250) — compile-verified
//
#include <hip/hip_runtime.h>
#include <hip/hip_bf16.h>
#include <math.h>

typedef __attribute__((ext_vector_type(2))) float v2f;
typedef __attribute__((ext_vector_type(8))) float v8f;
typedef unsigned int u32x4 __attribute__((ext_vector_type(4)));
typedef int i32x4 __attribute__((ext_vector_type(4)));
typedef int i32x8 __attribute__((ext_vector_type(8)));

#define B_ 512
#define S_ 1024
#define D_ 64
#define H_ 64
#define SD_ 4
#define K1_WAVES 8

#if defined(__has_builtin)
#if __has_builtin(__builtin_amdgcn_tensor_load_to_lds) && \
    __has_builtin(__builtin_amdgcn_s_wait_tensorcnt)
#define HAVE_TDM 1
#endif
#endif

__device__ __forceinline__ float gelu_exact(float v) {
  return 0.5f * v * (1.0f + erff(v * 0.70710678118654752440f));
}
__device__ __forceinline__ float sigmoidf_(float v) {
  return 1.0f / (1.0f + expf(-v));
}

// ---------------------------------------------------------------------------
// Kernel 1: h = GELU(LN(x @ W1^T + b1)), bx = h @ Wi^T + bi, for all B*S rows.
// One wave per 16-row tile. fp32 WMMA 16x16x4, K=64 -> 16 k-chunks, N=64 -> 4 tiles.
// ---------------------------------------------------------------------------
__global__ __launch_bounds__(256)
void proj_ln_gelu_kernel(const float* __restrict__ x,
                         const float* __restrict__ W1,
                         const float* __restrict__ b1,
                         const float* __restrict__ ln_g,
                         const float* __restrict__ ln_b,
                         const float* __restrict__ Wi,
                         const float* __restrict__ bi,
                         float* __restrict__ h_out,
                         float* __restrict__ bx_out)
{
  __shared__ float sW1[64 * 68];            // W1[h][d], pitch 68 (conflict-free)
  __shared__ float sX[K1_WAVES][16 * 68];   // per-wave x tile, pitch 68
  __shared__ float sWi[4 * 64];
  __shared__ float sB1[64], sG[64], sBt[64], sBi[4];

  const int tid  = threadIdx.x;
  const int wave = tid >> 5;
  const int lane = tid & 31;
  const int l15  = lane & 15;
  const int half = lane >> 4;
  const int koff = half * 2;

  for (int i = tid; i < 64 * 64; i += 256) sW1[(i >> 6) * 68 + (i & 63)] = W1[i];
  for (int i = tid; i < 4 * 64; i += 256)  sWi[i] = Wi[i];
  if (tid < 64) { sB1[tid] = b1[tid]; sG[tid] = ln_g[tid]; sBt[tid] = ln_b[tid]; }
  if (tid < 4)  sBi[tid] = bi[tid];

  const int tile = blockIdx.x * K1_WAVES + wave;     // 0..32767 row-tiles
  const float* xrow = x + (size_t)tile * (16 * 64);
  float* sXw = &sX[wave][0];
  for (int i = lane; i < 256; i += 32) {
    int r = i >> 4, c4 = i & 15;
    ((float4*)(sXw + r * 68))[c4] = ((const float4*)(xrow + r * 64))[c4];
  }
  __syncthreads();

  v2f a[16];
#pragma unroll
  for (int kc = 0; kc < 16; ++kc) {
    int k = kc * 4 + koff;
    a[kc].x = sXw[l15 * 68 + k];
    a[kc].y = sXw[l15 * 68 + k + 1];
  }

  v8f acc[4];
#pragma unroll
  for (int nt = 0; nt < 4; ++nt) {
    v8f c = {};
    const float* wb = sW1 + (nt * 16 + l15) * 68;    // B[k][n] = W1[n][k]
#pragma unroll
    for (int kc = 0; kc < 16; ++kc) {
      int k = kc * 4 + koff;
      v2f bf; bf.x = wb[k]; bf.y = wb[k + 1];
      c = __builtin_amdgcn_wmma_f32_16x16x4_f32(false, a[kc], false, bf,
                                                (short)0, c, false, false);
    }
    acc[nt] = c;
  }

  float s1[8], s2[8];
#pragma unroll
  for (int r = 0; r < 8; ++r) { s1[r] = 0.f; s2[r] = 0.f; }
#pragma unroll
  for (int nt = 0; nt < 4; ++nt) {
    float bias = sB1[nt * 16 + l15];
#pragma unroll
    for (int r = 0; r < 8; ++r) {
      float v = acc[nt][r] + bias;
      acc[nt][r] = v;
      s1[r] += v;
      s2[r] += v * v;
    }
  }
  for (int off = 1; off < 16; off <<= 1) {
#pragma unroll
    for (int r = 0; r < 8; ++r) {
      s1[r] += __shfl_xor(s1[r], off, 32);
      s2[r] += __shfl_xor(s2[r], off, 32);
    }
  }

  float p0[8], p1[8], p2[8], p3[8];
#pragma unroll
  for (int r = 0; r < 8; ++r) { p0[r] = p1[r] = p2[r] = p3[r] = 0.f; }

#pragma unroll
  for (int nt = 0; nt < 4; ++nt) {
    int n = nt * 16 + l15;
    float g = sG[n], bt = sBt[n];
    float w0 = sWi[0 * 64 + n], w1 = sWi[1 * 64 + n];
    float w2 = sWi[2 * 64 + n], w3 = sWi[3 * 64 + n];
#pragma unroll
    for (int r = 0; r < 8; ++r) {
      float mu   = s1[r] * (1.0f / 64.0f);
      float var  = s2[r] * (1.0f / 64.0f) - mu * mu;
      float rstd = rsqrtf(var + 1e-5f);
      float hv = gelu_exact((acc[nt][r] - mu) * rstd * g + bt);
      p0[r] += hv * w0; p1[r] += hv * w1; p2[r] += hv * w2; p3[r] += hv * w3;
      int m = r + half * 8;
      h_out[((size_t)tile * 16 + m) * 64 + n] = hv;
    }
  }
  for (int off = 1; off < 16; off <<= 1) {
#pragma unroll
    for (int r = 0; r < 8; ++r) {
      p0[r] += __shfl_xor(p0[r], off, 32);
      p1[r] += __shfl_xor(p1[r], off, 32);
      p2[r] += __shfl_xor(p2[r], off, 32);
      p3[r] += __shfl_xor(p3[r], off, 32);
    }
  }
  if (l15 < 4) {
#pragma unroll
    for (int r = 0; r < 8; ++r) {
      int m = r + half * 8;
      float v = (l15 == 0) ? p0[r] : (l15 == 1) ? p1[r] : (l15 == 2) ? p2[r] : p3[r];
      bx_out[((size_t)tile * 16 + m) * 4 + l15] = v + sBi[l15];
    }
  }
}

// ---------------------------------------------------------------------------
// Kernel 2: sequential scan. One block per 16 batch rows (32 blocks x 128 thr).
// h tile for step t+1 is DMA'd by the Tensor Data Mover into the ping-pong LDS
// buffer while step t computes (17+16 fp32 WMMAs). TDM descriptor: 2D tile
// 64x16 elem, row stride S*64 elem, pad 12 DW per 64 DW -> LDS pitch 76 floats
// landing at column 4 of sCorr (cols 0..3 hold s_lin).
// ---------------------------------------------------------------------------
#ifdef HAVE_TDM
__device__ __forceinline__ void tdm_load_h_tile(const float* gsrc,
                                                unsigned int lds_byte_addr) {
  unsigned long long gaddr = (unsigned long long)(uintptr_t)gsrc;
  u32x4 g0;
  g0.x = 1u;                                   // count=1, user descriptor
  g0.y = lds_byte_addr;                        // LDS dest (bytes)
  g0.z = (unsigned int)gaddr;                  // global_addr[31:0]
  g0.w = (unsigned int)((gaddr >> 32) & 0x01FFFFFFu) | (2u << 30); // [56:32]|type=2
  i32x8 g1;
  g1[0] = (2 << 16)      // data_size = 4 bytes
        | (1 << 20)      // pad_enable
        | (5 << 22)      // pad_interval: 64 DWORDs
        | (11 << 25);    // pad_amount: 12 DWORDs  (=> pitch 76 floats)
  g1[1] = (64 << 16);    // tensor_dim0 = 64 (bits 79:48)
  g1[2] = 0;             // tensor_dim0 hi / tensor_dim1 lo16 (=0)
  g1[3] = (64 << 16) | ((B_ * S_) >> 16);   // tile_dim0=64 | tensor_dim1[31:16]
  g1[4] = 16;            // tile_dim1 = 16 rows, tile_dim2 = 0
  g1[5] = S_ * 64;       // tensor_dim0_stride[31:0] (elements between rows)
  g1[6] = 0;
  g1[7] = 0;
  i32x4 g2 = {0, 0, 0, 0};
  i32x4 g3 = {0, 0, 0, 0};
#if __clang_major__ >= 23
  i32x8 g4 = {0, 0, 0, 0, 0, 0, 0, 0};
  __builtin_amdgcn_tensor_load_to_lds(g0, g1, g2, g3, g4, 0);
#else
  __builtin_amdgcn_tensor_load_to_lds(g0, g1, g2, g3, 0);
#endif
}
#endif

__global__ __launch_bounds__(128)
void scan_kernel(const float* __restrict__ h_ws,
                 const float* __restrict__ bx_ws,
                 const float* __restrict__ Wc1,
                 const float* __restrict__ bc1,
                 const float* __restrict__ Wc2,
                 const float* __restrict__ bc2,
                 const float* __restrict__ corr_scale,
                 const float* __restrict__ A_level,
                 const float* __restrict__ A_trend,
                 const float* __restrict__ A_gamma,
                 const float* __restrict__ A_resid,
                 const float* __restrict__ omega,
                 float* __restrict__ out)
{
  __shared__ float sCorr[2][16 * 76];  // ping-pong: cols 0..3 s_lin, 4..67 h_t
  __shared__ float sHid[16 * 68];
  __shared__ float sS[16 * 4];

  const int tid  = threadIdx.x;
  const int wave = tid >> 5;
  const int lane = tid & 31;
  const int l15  = lane & 15;
  const int half = lane >> 4;
  const int koff = half * 2;
  const int b0   = blockIdx.x * 16;

  const float a0 = sigmoidf_(A_level[0]) * 0.15f + 0.85f;
  const float a1 = sigmoidf_(A_trend[0]) * 0.25f + 0.70f;
  const float a2 = (sigmoidf_(A_gamma[0]) * 0.20f + 0.80f) * cosf(omega[0]);
  const float a3 = sigmoidf_(A_resid[0]) * 0.40f;
  const float cs = corr_scale[0];

  // GEMM1 B fragments (Wc1^T, K=68), pinned in registers for the whole scan.
  v2f bw1[17];
  {
    const float* wrow = Wc1 + (size_t)(wave * 16 + l15) * 68;
#pragma unroll
    for (int kc = 0; kc < 17; ++kc) {
      int k = kc * 4 + koff;
      bw1[kc].x = wrow[k];
      bw1[kc].y = wrow[k + 1];
    }
  }
  const float bc1v = bc1[wave * 16 + l15];

  // GEMM2 B fragments (Wc2^T, N=0..3 valid), used by wave 0.
  v2f bw2[16];
#pragma unroll
  for (int kc = 0; kc < 16; ++kc) {
    int k = kc * 4 + koff;
    if (l15 < 4) { bw2[kc].x = Wc2[l15 * 64 + k]; bw2[kc].y = Wc2[l15 * 64 + k + 1]; }
    else         { bw2[kc].x = 0.f;               bw2[kc].y = 0.f; }
  }
  const float bc2v = (l15 < 4) ? bc2[l15] : 0.f;

  for (int i = tid; i < 64; i += 128) sS[i] = 0.f;

#ifdef HAVE_TDM
  // Prologue: DMA h tile for t=0 into buffer 0 (column offset 4 -> +16 bytes).
  if (wave == 0) {
    unsigned int lds0 = (unsigned int)(uintptr_t)(&sCorr[0][0]) + 16u;
    tdm_load_h_tile(h_ws + ((size_t)b0 * S_ + 0) * 64, lds0);
  }
#endif
  __syncthreads();

  for (int t = 0; t < S_; ++t) {
    const int buf = t & 1;
    float* sC = &sCorr[buf][0];

#ifdef HAVE_TDM
    if (wave == 0) __builtin_amdgcn_s_wait_tensorcnt(0);  // h(t) landed in LDS
    __syncthreads();                                      // publish to all waves
    if (wave == 0 && t + 1 < S_) {                        // overlap: DMA h(t+1)
      unsigned int ldsN = (unsigned int)(uintptr_t)(&sCorr[buf ^ 1][0]) + 16u;
      tdm_load_h_tile(h_ws + ((size_t)b0 * S_ + (t + 1)) * 64, ldsN);
    }
#else
    {
      int row = tid >> 3;
      int c8  = (tid & 7) * 8;
      const float* hp = h_ws + ((size_t)(b0 + row) * S_ + t) * 64 + c8;
      float4 v0 = ((const float4*)hp)[0];
      float4 v1 = ((const float4*)hp)[1];
      float* dst = &sC[row * 76 + 4 + c8];
      dst[0] = v0.x; dst[1] = v0.y; dst[2] = v0.z; dst[3] = v0.w;
      dst[4] = v1.x; dst[5] = v1.y; dst[6] = v1.z; dst[7] = v1.w;
    }
#endif
    // s_lin = diag(A)*s + bx  ->  cols 0..3
    if (tid < 64) {
      int row = tid >> 2, k = tid & 3;
      float ak  = (k == 0) ? a0 : (k == 1) ? a1 : (k == 2) ? a2 : a3;
      float bxv = bx_ws[((size_t)(b0 + row) * S_ + t) * 4 + k];
      sC[row * 76 + k] = ak * sS[row * 4 + k] + bxv;
      if (t + 1 < S_)
        __builtin_prefetch(bx_ws + ((size_t)(b0 + row) * S_ + (t + 1)) * 4, 0, 0);
    }
    __syncthreads();

    // GEMM1: corr_in[16x68] @ Wc1^T -> this wave's 16x16 hidden tile
    v8f acc = {};
#pragma unroll
    for (int kc = 0; kc < 17; ++kc) {
      int k = kc * 4 + koff;
      v2f av;
      av.x = sC[l15 * 76 + k];
      av.y = sC[l15 * 76 + k + 1];
      acc = __builtin_amdgcn_wmma_f32_16x16x4_f32(false, av, false, bw1[kc],
                                                  (short)0, acc, false, false);
    }
#pragma unroll
    for (int r = 0; r < 8; ++r) {
      int m = r + half * 8;
      sHid[m * 68 + wave * 16 + l15] = gelu_exact(acc[r] + bc1v);
    }
    __syncthreads();

    // GEMM2 on wave 0: hidden[16x64] @ Wc2^T -> c[16x4]; state update
    if (wave == 0) {
      v8f acc2 = {};
#pragma unroll
      for (int kc = 0; kc < 16; ++kc) {
        int k = kc * 4 + koff;
        v2f av;
        av.x = sHid[l15 * 68 + k];
        av.y = sHid[l15 * 68 + k + 1];
        acc2 = __builtin_amdgcn_wmma_f32_16x16x4_f32(false, av, false, bw2[kc],
                                                     (short)0, acc2, false, false);
      }
      if (l15 < 4) {
#pragma unroll
        for (int r = 0; r < 8; ++r) {
          int m = r + half * 8;
          float slin = sC[m * 76 + l15];
          float snew = slin + cs * tanhf(acc2[r] + bc2v);
          sS[m * 4 + l15] = snew;
          out[((size_t)(b0 + m) * S_ + t) * 4 + l15] = snew;
        }
      }
    }
    __syncthreads();
  }
}

// ---------------------------------------------------------------------------
extern "C" void kernel_launch(void* const* d_in, const int* in_sizes, int n_in,
                              void* d_out, int out_size, void* d_ws, size_t ws_size,
                              hipStream_t stream) {
  (void)in_sizes; (void)n_in; (void)out_size; (void)ws_size;
  const float* x       = (const float*)d_in[0];
  const float* W1      = (const float*)d_in[1];
  const float* b1      = (const float*)d_in[2];
  const float* ln_g    = (const float*)d_in[3];
  const float* ln_b    = (const float*)d_in[4];
  const float* Wi      = (const float*)d_in[5];
  const float* bi      = (const float*)d_in[6];
  const float* Wc1     = (const float*)d_in[7];
  const float* bc1     = (const float*)d_in[8];
  const float* Wc2     = (const float*)d_in[9];
  const float* bc2     = (const float*)d_in[10];
  const float* cscal   = (const float*)d_in[11];
  const float* A_level = (const float*)d_in[12];
  const float* A_trend = (const float*)d_in[13];
  const float* A_gamma = (const float*)d_in[14];
  const float* A_resid = (const float*)d_in[15];
  const float* omega   = (const float*)d_in[16];

  float* h_ws  = (float*)d_ws;                       // [B*S, 64] = 134 MB
  float* bx_ws = h_ws + (size_t)B_ * S_ * H_;        // [B*S, 4]  = 8 MB

  dim3 g1(B_ * S_ / 16 / K1_WAVES);                  // 4096 blocks x 256 thr
  proj_ln_gelu_kernel<<<g1, 256, 0, stream>>>(x, W1, b1, ln_g, ln_b, Wi, bi,
                                              h_ws, bx_ws);

  dim3 g2(B_ / 16);                                  // 32 blocks x 128 thr
  scan_kernel<<<g2, 128, 0, stream>>>(h_ws, bx_ws, Wc1, bc1, Wc2, bc2, cscal,
                                      A_level, A_trend, A_gamma, A_resid, omega,
                                      (float*)d_out);
}